// RtConditionedPredictor_32830730010783
// MI455X (gfx1250) — compile-verified
//
#include <hip/hip_runtime.h>
#include <math.h>

typedef __attribute__((ext_vector_type(2))) float v2f;
typedef __attribute__((ext_vector_type(8))) float v8f;

#define BB 256
#define NNODES 2048
#define DDIM 32
#define HOR 7
#define MROWS (BB * NNODES)          // 524288
#define ROWS_PER_BLOCK 128
#define YS 132                        // padded row stride for Y buffer (128 cols)
#define HS 56                         // padded row stride for H3 buffer (55 cols)

// ---- LDS layout (float offsets) ----
#define OFF_WF   0                    // 8192  : W1 fused, pre-swizzled to B-fragment order
#define OFF_YS   8192                 // 16896 : 128 x 132 activation buffer
#define OFF_H3   25088                // 7168  : 128 x 56 layer-2 outputs + refine logits
#define OFF_HB1  32256                // 96
#define OFF_HG   32352                // 96
#define OFF_HBE  32448                // 96
#define OFF_HW2  32544                // 1536
#define OFF_HB2  34080                // 48
#define OFF_HW3  34128                // 336
#define OFF_HB3  34464                // 21
#define OFF_GW1  34485                // 16
#define OFF_GB1  34501                // 16
#define OFF_GW2  34517                // 48
#define OFF_GB2  34565                // 3
#define OFF_RB1  34568                // 32
#define OFF_RW2  34600                // 224
#define OFF_RB2  34824                // 7
#define LDS_FLOATS 34831              // ~136 KB dynamic LDS

__device__ __forceinline__ v8f wmma4(v2f a, v2f b, v8f c) {
    // V_WMMA_F32_16X16X4_F32 : full fp32 matrix op (no precision downconvert)
    return __builtin_amdgcn_wmma_f32_16x16x4_f32(
        false, a, false, b, (short)0, c, false, false);
}

__global__ __launch_bounds__(256)
void rtpred_fused_kernel(const float* __restrict__ features,
                         const float* __restrict__ rt_emb,
                         const float* __restrict__ rt_est,
                         const float* __restrict__ last_vals,
                         const float* __restrict__ hw1, const float* __restrict__ hb1,
                         const float* __restrict__ hg,  const float* __restrict__ hbe,
                         const float* __restrict__ hw2, const float* __restrict__ hb2,
                         const float* __restrict__ hw3, const float* __restrict__ hb3,
                         const float* __restrict__ gw1, const float* __restrict__ gb1,
                         const float* __restrict__ gw2, const float* __restrict__ gb2,
                         const float* __restrict__ rw1, const float* __restrict__ rb1,
                         const float* __restrict__ rw2, const float* __restrict__ rb2,
                         float* __restrict__ out)
{
    extern __shared__ float smem[];
    const int tid  = threadIdx.x;
    const int lane = tid & 31;
    const int wave = tid >> 5;
    const int m    = lane & 15;       // A-matrix row within tile / B col
    const int half = lane >> 4;       // 0 or 1
    const int koff = half << 1;       // K sub-offset per ISA 16x16x4 layouts
    const int n    = lane & 15;       // output column within tile
    const int grow0 = blockIdx.x * ROWS_PER_BLOCK;

    // ================= stage weights into LDS =================
    // W1 fused [64 x 128] (cols 0..95 = hw1 heads, 96..127 = rw1),
    // stored directly in WMMA B-fragment order: frag = j*8 + t (kstep, coltile),
    // element layout = lane-major pairs -> conflict-free ds_load_b64 in the hot loop.
    {
        float* Wf = smem + OFF_WF;
        for (int i = tid; i < 8192; i += 256) {
            int frag   = i >> 6;              // 0..127
            int within = i & 63;
            int l      = within >> 1;         // lane 0..31
            int comp   = within & 1;          // .x / .y
            int j      = frag >> 3;           // kstep 0..15
            int t      = frag & 7;            // col tile 0..7
            int srow   = 4 * j + ((l >> 4) << 1) + comp;   // K index 0..63
            int scol   = 16 * t + (l & 15);                // col  0..127
            float v = (scol < 96)
                          ? hw1[(scol >> 5) * 2048 + srow * 32 + (scol & 31)]
                          : rw1[srow * 32 + (scol - 96)];
            Wf[i] = v;
        }
    }
    for (int i = tid; i < 96; i += 256) {
        smem[OFF_HB1 + i] = hb1[i];
        smem[OFF_HG  + i] = hg[i];
        smem[OFF_HBE + i] = hbe[i];
    }
    for (int i = tid; i < 1536; i += 256) smem[OFF_HW2 + i] = hw2[i];
    for (int i = tid; i < 336;  i += 256) smem[OFF_HW3 + i] = hw3[i];
    for (int i = tid; i < 224;  i += 256) smem[OFF_RW2 + i] = rw2[i];
    if (tid < 48) { smem[OFF_HB2 + tid] = hb2[tid]; smem[OFF_GW2 + tid] = gw2[tid]; }
    if (tid < 21) smem[OFF_HB3 + tid] = hb3[tid];
    if (tid < 16) { smem[OFF_GW1 + tid] = gw1[tid]; smem[OFF_GB1 + tid] = gb1[tid]; }
    if (tid < 32) smem[OFF_RB1 + tid] = rb1[tid];
    if (tid < 3)  smem[OFF_GB2 + tid] = gb2[tid];
    if (tid < 7)  smem[OFF_RB2 + tid] = rb2[tid];
    __syncthreads();

    float* Ys = smem + OFF_YS;
    float* H3 = smem + OFF_H3;

    // ================= phase 1: Y = combined @ W1fused  (WMMA, K=64, 128 cols) ====
    {
        const int rowg = grow0 + wave * 16 + m;
        const float* frow = features + (size_t)rowg * DDIM;
        const float* erow = rt_emb   + (size_t)rowg * DDIM;
        v2f afrag[16];
        #pragma unroll
        for (int j = 0; j < 16; ++j) {
            int k = 4 * j + koff;
            const float* src = (j < 8) ? (frow + k) : (erow + (k - 32));
            afrag[j] = *(const v2f*)src;
        }
        const v2f* Wf2 = (const v2f*)(smem + OFF_WF);
        #pragma unroll
        for (int t = 0; t < 8; ++t) {
            v8f acc = {};
            #pragma unroll
            for (int j = 0; j < 16; ++j) {
                v2f b = Wf2[(j * 8 + t) * 32 + lane];
                acc = wmma4(afrag[j], b, acc);
            }
            int rbase = wave * 16 + (half << 3);
            #pragma unroll
            for (int r = 0; r < 8; ++r)
                Ys[(rbase + r) * YS + 16 * t + n] = acc[r];
        }
    }
    __syncthreads();

    // ================= phase 2: bias + LayerNorm + ReLU (scalar, in-place) ========
    #pragma unroll
    for (int it = 0; it < 2; ++it) {
        int task = tid + it * 256;        // 512 tasks: 128 rows x {head0,1,2,refine}
        int row  = task & 127;
        int part = task >> 7;             // 0..3
        float* yr = Ys + row * YS;
        if (part < 3) {
            int cb = part * 32;
            float v[32];
            float mu = 0.f;
            #pragma unroll
            for (int i = 0; i < 32; ++i) { v[i] = yr[cb + i] + smem[OFF_HB1 + cb + i]; mu += v[i]; }
            mu *= 0.03125f;
            float var = 0.f;
            #pragma unroll
            for (int i = 0; i < 32; ++i) { float d = v[i] - mu; var += d * d; }
            var *= 0.03125f;
            float inv = rsqrtf(var + 1e-5f);
            #pragma unroll
            for (int i = 0; i < 32; ++i) {
                float hn = (v[i] - mu) * inv * smem[OFF_HG + cb + i] + smem[OFF_HBE + cb + i];
                yr[cb + i] = hn > 0.f ? hn : 0.f;
            }
        } else {
            #pragma unroll
            for (int i = 0; i < 32; ++i) {
                float z = yr[96 + i] + smem[OFF_RB1 + i];
                yr[96 + i] = z > 0.f ? z : 0.f;
            }
        }
    }
    __syncthreads();

    // ================= phase 3: layer-2 (WMMA, K=32) per head + refine [32->7] =====
    {
        const int nn = (n < 7) ? n : 0;
        const int rrow = wave * 16 + m;
        #pragma unroll
        for (int hidx = 0; hidx < 4; ++hidx) {     // 0..2 heads, 3 = refine
            int cb = 32 * hidx;
            v8f acc = {};
            #pragma unroll
            for (int j = 0; j < 8; ++j) {
                v2f a = *(const v2f*)&Ys[rrow * YS + cb + 4 * j + koff];
                int kr = 4 * j + koff;
                v2f b;
                if (hidx < 3) {
                    b.x = smem[OFF_HW2 + hidx * 512 + kr * 16 + n];
                    b.y = smem[OFF_HW2 + hidx * 512 + (kr + 1) * 16 + n];
                } else {
                    float b0 = smem[OFF_RW2 + kr * 7 + nn];
                    float b1 = smem[OFF_RW2 + (kr + 1) * 7 + nn];
                    b.x = (n < 7) ? b0 : 0.f;      // value select, no EXEC change
                    b.y = (n < 7) ? b1 : 0.f;
                }
                acc = wmma4(a, b, acc);
            }
            int rbase = wave * 16 + (half << 3);
            if (hidx < 3) {
                float bias = smem[OFF_HB2 + hidx * 16 + n];
                #pragma unroll
                for (int r = 0; r < 8; ++r) {
                    float v = acc[r] + bias;
                    H3[(rbase + r) * HS + hidx * 16 + n] = v > 0.f ? v : 0.f;
                }
            } else if (n < 7) {
                float bias = smem[OFF_RB2 + n];
                #pragma unroll
                for (int r = 0; r < 8; ++r)
                    H3[(rbase + r) * HS + 48 + n] = acc[r] + bias;   // refine logits
            }
        }
    }
    __syncthreads();

    // ================= phase 4: layer-3 (WMMA, K=16) per head [16->7] =============
    {
        const int nn = (n < 7) ? n : 0;
        const int rrow = wave * 16 + m;
        #pragma unroll
        for (int hk = 0; hk < 3; ++hk) {
            v8f acc = {};
            #pragma unroll
            for (int j = 0; j < 4; ++j) {
                v2f a = *(const v2f*)&H3[rrow * HS + hk * 16 + 4 * j + koff];
                int kr = 4 * j + koff;
                float b0 = smem[OFF_HW3 + hk * 112 + kr * 7 + nn];
                float b1 = smem[OFF_HW3 + hk * 112 + (kr + 1) * 7 + nn];
                v2f b;
                b.x = (n < 7) ? b0 : 0.f;
                b.y = (n < 7) ? b1 : 0.f;
                acc = wmma4(a, b, acc);
            }
            if (n < 7) {
                float bias = smem[OFF_HB3 + hk * 7 + n];
                int rbase = wave * 16 + (half << 3);
                #pragma unroll
                for (int r = 0; r < 8; ++r)
                    Ys[(rbase + r) * YS + hk * 7 + n] = acc[r] + bias;  // preds -> Ys cols 0..20
            }
        }
    }
    __syncthreads();

    // ================= phase 5: gate softmax + sigmoid refine blend ===============
    if (tid < ROWS_PER_BLOCK) {
        const int row  = tid;
        const int grow = grow0 + row;
        float rt = rt_est[grow];
        float g[16];
        #pragma unroll
        for (int i = 0; i < 16; ++i) {
            float v = rt * smem[OFF_GW1 + i] + smem[OFF_GB1 + i];
            g[i] = v > 0.f ? v : 0.f;
        }
        float logit[3];
        #pragma unroll
        for (int k = 0; k < 3; ++k) {
            float s = smem[OFF_GB2 + k];
            #pragma unroll
            for (int i = 0; i < 16; ++i) s += g[i] * smem[OFF_GW2 + i * 3 + k];
            logit[k] = s;
        }
        float mx = fmaxf(logit[0], fmaxf(logit[1], logit[2]));
        float e0 = expf(logit[0] - mx), e1 = expf(logit[1] - mx), e2 = expf(logit[2] - mx);
        float isum = 1.f / (e0 + e1 + e2);
        float g0 = e0 * isum, g1 = e1 * isum, g2 = e2 * isum;
        float lv = last_vals[grow];
        const float* yr = Ys + row * YS;
        const float* hr = H3 + row * HS;
        #pragma unroll
        for (int o = 0; o < 7; ++o) {
            float w   = g0 * yr[o] + g1 * yr[7 + o] + g2 * yr[14 + o];
            float ref = 1.f / (1.f + expf(-hr[48 + o]));
            out[(size_t)grow * 7 + o] = ref * w + (1.f - ref) * lv;
        }
    }
}

extern "C" void kernel_launch(void* const* d_in, const int* in_sizes, int n_in,
                              void* d_out, int out_size, void* d_ws, size_t ws_size,
                              hipStream_t stream) {
    (void)in_sizes; (void)n_in; (void)out_size; (void)d_ws; (void)ws_size;
    const float* features = (const float*)d_in[0];
    const float* rt_emb   = (const float*)d_in[1];
    const float* rt_est   = (const float*)d_in[2];
    const float* last_v   = (const float*)d_in[3];
    const float* hw1      = (const float*)d_in[4];
    const float* hb1      = (const float*)d_in[5];
    const float* hg       = (const float*)d_in[6];
    const float* hbe      = (const float*)d_in[7];
    const float* hw2      = (const float*)d_in[8];
    const float* hb2      = (const float*)d_in[9];
    const float* hw3      = (const float*)d_in[10];
    const float* hb3      = (const float*)d_in[11];
    const float* gw1      = (const float*)d_in[12];
    const float* gb1      = (const float*)d_in[13];
    const float* gw2      = (const float*)d_in[14];
    const float* gb2      = (const float*)d_in[15];
    const float* rw1      = (const float*)d_in[16];
    const float* rb1      = (const float*)d_in[17];
    const float* rw2      = (const float*)d_in[18];
    const float* rb2      = (const float*)d_in[19];

    dim3 grid(MROWS / ROWS_PER_BLOCK);   // 4096 blocks
    dim3 block(256);                     // 8 wave32s
    size_t shmem = (size_t)LDS_FLOATS * sizeof(float);  // ~136 KB dynamic LDS

    rtpred_fused_kernel<<<grid, block, shmem, stream>>>(
        features, rt_emb, rt_est, last_v,
        hw1, hb1, hg, hbe, hw2, hb2, hw3, hb3,
        gw1, gb1, gw2, gb2, rw1, rb1, rw2, rb2,
        (float*)d_out);
}